// Meta_74010876445283
// MI455X (gfx1250) — compile-verified
//
#include <hip/hip_runtime.h>
#include <hip/hip_bf16.h>

typedef __attribute__((ext_vector_type(2))) float v2f;
typedef __attribute__((ext_vector_type(8))) float v8f;

#define NROW   153600       // N*M*T
#define DIM    256
#define KCLS   5
#define NN     256
#define MM     2
#define TT     300
#define TOPK   64
#define EPSF   1e-8f

// workspace layout (float indices)
#define WS_SUM    0          // 5*256 class column sums
#define WS_CNT    1280       // 8 ints (class counts)
#define WS_MU     1288       // 16*256 zero-padded class means
#define WS_INTER  5384       // 16
#define WS_MUSQ   5400       // 16
#define WS_SB     5416       // 1
#define WS_SW     5417       // 1
#define WS_SCORE  5424       // NROW frame scores
#define WS_ZERON  5424       // floats to zero at start

// ---------------- kernel 0: zero accumulators ----------------
__global__ void k_zero(float* ws) {
    int i = blockIdx.x * blockDim.x + threadIdx.x;
    if (i < WS_ZERON) ws[i] = 0.0f;
}

// ---------------- kernel 1: per-class column sums + counts ----------------
__global__ __launch_bounds__(256) void k_class_sums(const float* __restrict__ xT,
                                                    const int* __restrict__ lab,
                                                    float* ws) {
    const int t = threadIdx.x;                       // column [0,256)
    const int rowsPerBlock = (NROW + gridDim.x - 1) / gridDim.x;
    const int r0 = blockIdx.x * rowsPerBlock;
    const int r1 = (r0 + rowsPerBlock < NROW) ? (r0 + rowsPerBlock) : NROW;

    float a0 = 0.f, a1 = 0.f, a2 = 0.f, a3 = 0.f, a4 = 0.f;
    for (int r = r0; r < r1; ++r) {
        const int l = lab[r];                        // scalar broadcast load
        const float v = xT[(size_t)r * DIM + t];     // coalesced 1KB per row
        a0 += (l == 0) ? v : 0.f;
        a1 += (l == 1) ? v : 0.f;
        a2 += (l == 2) ? v : 0.f;
        a3 += (l == 3) ? v : 0.f;
        a4 += (l == 4) ? v : 0.f;
    }
    atomicAdd(&ws[WS_SUM + 0 * DIM + t], a0);
    atomicAdd(&ws[WS_SUM + 1 * DIM + t], a1);
    atomicAdd(&ws[WS_SUM + 2 * DIM + t], a2);
    atomicAdd(&ws[WS_SUM + 3 * DIM + t], a3);
    atomicAdd(&ws[WS_SUM + 4 * DIM + t], a4);

    __shared__ int cnt[KCLS];
    if (t < KCLS) cnt[t] = 0;
    __syncthreads();
    for (int r = r0 + t; r < r1; r += blockDim.x) atomicAdd(&cnt[lab[r]], 1);
    __syncthreads();
    int* wsi = (int*)ws;
    if (t < KCLS) atomicAdd(&wsi[WS_CNT + t], cnt[t]);
}

// ---------------- kernel 2: class means / inter / musq / Sb (1 block) ----------------
__global__ __launch_bounds__(256) void k_stats(float* ws) {
    const int t = threadIdx.x;                       // column
    int* wsi = (int*)ws;
    __shared__ float redA[DIM], redB[DIM];
    __shared__ float interS[KCLS];

    float sums[KCLS];
    float tot = 0.f;
#pragma unroll
    for (int c = 0; c < KCLS; ++c) { sums[c] = ws[WS_SUM + c * DIM + t]; tot += sums[c]; }
    const float om = tot / (float)NROW;              // overall mean, column t

    // zero-pad mean rows 5..15 (columns 5..15 of B in the WMMA GEMM)
    for (int c = KCLS; c < 16; ++c) ws[WS_MU + c * DIM + t] = 0.f;

    for (int c = 0; c < KCLS; ++c) {
        const float cn = (float)wsi[WS_CNT + c];
        const float mean = sums[c] / fmaxf(cn, 1.f);
        ws[WS_MU + c * DIM + t] = mean;
        const float d = mean - om;
        redA[t] = d * d;
        redB[t] = mean * mean;
        __syncthreads();
        for (int s = DIM / 2; s > 0; s >>= 1) {
            if (t < s) { redA[t] += redA[t + s]; redB[t] += redB[t + s]; }
            __syncthreads();
        }
        if (t == 0) {
            interS[c] = redA[0];
            ws[WS_INTER + c] = redA[0];
            ws[WS_MUSQ + c] = redB[0];
        }
        __syncthreads();
    }
    if (t >= KCLS && t < 16) { ws[WS_INTER + t] = 0.f; ws[WS_MUSQ + t] = 0.f; }
    if (t == 0) {
        float sb = 0.f;
        for (int c = 0; c < KCLS; ++c) sb += (float)wsi[WS_CNT + c] * interS[c];
        ws[WS_SB] = sb;
    }
}

// ---------------- kernel 3: WMMA intra-distance + scores + Sw ----------------
// D = X(16x256) @ MuPad^T(256x16) via V_WMMA_F32_16X16X4_F32, K-loop of 64.
__global__ __launch_bounds__(256) void k_intra(const float* __restrict__ xT,
                                               const int* __restrict__ lab,
                                               float* ws) {
    __shared__ float muS[16 * DIM];      // 16 KB: zero-padded class means (B operand)
    __shared__ float dS[8 * 256];        // 8 KB: per-wave 16x16 accumulator spill
    __shared__ float interS[16], musqS[16];

    for (int i = threadIdx.x; i < 16 * DIM; i += blockDim.x) muS[i] = ws[WS_MU + i];
    if (threadIdx.x < 16) {
        interS[threadIdx.x] = ws[WS_INTER + threadIdx.x];
        musqS[threadIdx.x]  = ws[WS_MUSQ + threadIdx.x];
    }
    __syncthreads();

    const int lane  = threadIdx.x & 31;
    const int wib   = threadIdx.x >> 5;
    const int wave  = blockIdx.x * (blockDim.x >> 5) + wib;
    const int nwave = gridDim.x * (blockDim.x >> 5);
    const int m  = lane & 15;            // A: row (M) / B: column (N)
    const int kh = lane >> 4;            // K half: lanes 0-15 -> K{0,1}, 16-31 -> K{2,3}
    const int ntiles = NROW / 16;

    for (int tile = wave; tile < ntiles; tile += nwave) {
        const int row0 = tile * 16;
        const float* xr = xT + (size_t)(row0 + m) * DIM + 2 * kh;
        const float* mr = muS + m * DIM + 2 * kh;

        v8f acc = {0.f, 0.f, 0.f, 0.f, 0.f, 0.f, 0.f, 0.f};
        float sq = 0.f;
#pragma unroll 4
        for (int k0 = 0; k0 < DIM; k0 += 4) {
            const float ax = xr[k0], ay = xr[k0 + 1];
            const float bx = mr[k0], by = mr[k0 + 1];
            v2f a = {ax, ay};
            v2f b = {bx, by};
            sq += ax * ax + ay * ay;
            acc = __builtin_amdgcn_wmma_f32_16x16x4_f32(
                false, a, false, b, (short)0, acc, false, false);
        }
        // full row norm: lanes L and L^16 hold complementary K halves of row L&15
        sq += __shfl_xor(sq, 16, 32);

        // spill D to LDS: element (VGPR j, lane p) = D[M = j + 8*(p>=16), N = p&15]
        float* dst = dS + wib * 256;
#pragma unroll
        for (int j = 0; j < 8; ++j) dst[j * 32 + lane] = acc[j];

        float swv = 0.f;
        if (lane < 16) {
            const int row = row0 + lane;
            const int l = lab[row];
            const float dot = dst[(lane & 7) * 32 + ((lane >= 8) ? 16 : 0) + l];
            const float it = sq - 2.f * dot + musqS[l];
            ws[WS_SCORE + row] = interS[l] / (it + EPSF);
            swv = it;
        }
        for (int off = 8; off > 0; off >>= 1) swv += __shfl_down(swv, off, 32);
        if (lane == 0) atomicAdd(&ws[WS_SW], swv);
    }
}

// ---------------- kernel 4: VF -> top-64 -> sorted idx -> gather + loss ----------------
__global__ __launch_bounds__(256) void k_topk_gather(const float* __restrict__ x,
                                                     float* __restrict__ out,
                                                     const float* __restrict__ ws) {
    const int n = blockIdx.x;
    const int tid = threadIdx.x;
    __shared__ float val[512];
    __shared__ int   idx[512];

    for (int i = tid; i < 512; i += 256) {
        float v = -INFINITY;
        if (i < TT) {
            const float s0 = ws[WS_SCORE + (size_t)(n * MM + 0) * TT + i];
            const float s1 = ws[WS_SCORE + (size_t)(n * MM + 1) * TT + i];
            v = 0.5f * (s0 + s1);
        }
        val[i] = v;
        idx[i] = i;
    }
    __syncthreads();

    // bitonic sort, descending by (value desc, index asc)
    for (int k = 2; k <= 512; k <<= 1) {
        for (int j = k >> 1; j > 0; j >>= 1) {
            for (int i = tid; i < 512; i += 256) {
                const int ixj = i ^ j;
                if (ixj > i) {
                    const float vi = val[i], vj = val[ixj];
                    const int ii = idx[i], ij = idx[ixj];
                    const bool iAfter = (vi < vj) || (vi == vj && ii > ij);
                    const bool doSwap = ((i & k) == 0) ? iAfter : !iAfter;
                    if (doSwap) { val[i] = vj; val[ixj] = vi; idx[i] = ij; idx[ixj] = ii; }
                }
            }
            __syncthreads();
        }
    }

    // sort the 64 selected indices ascending
    for (int k = 2; k <= TOPK; k <<= 1) {
        for (int j = k >> 1; j > 0; j >>= 1) {
            if (tid < TOPK) {
                const int i = tid, ixj = i ^ j;
                if (ixj > i) {
                    const int a = idx[i], b = idx[ixj];
                    const bool sw = ((i & k) == 0) ? (a > b) : (a < b);
                    if (sw) { idx[i] = b; idx[ixj] = a; }
                }
            }
            __syncthreads();
        }
    }

    // gather: x[n, c, idx[j], :, :] -> out[1 + ...], 50 contiguous floats per (c,j)
    for (int e = tid; e < 3 * TOPK * 50; e += 256) {
        const int c   = e / (TOPK * 50);
        const int rem = e % (TOPK * 50);
        const int jj  = rem / 50;
        const int vm  = rem % 50;
        const int t   = idx[jj];
        out[1 + (size_t)((n * 3 + c) * TOPK + jj) * 50 + vm] =
            x[(size_t)((n * 3 + c) * TT + t) * 50 + vm];
    }
    if (n == 0 && tid == 0) out[0] = ws[WS_SW] / (ws[WS_SB] + EPSF);
}

extern "C" void kernel_launch(void* const* d_in, const int* in_sizes, int n_in,
                              void* d_out, int out_size, void* d_ws, size_t ws_size,
                              hipStream_t stream) {
    const float* x   = (const float*)d_in[0];   // [256,3,300,25,2]
    const float* xT  = (const float*)d_in[1];   // [153600,256]
    const int*   lab = (const int*)d_in[2];     // [153600]
    float* out = (float*)d_out;                 // [1 + 256*3*64*25*2]
    float* ws  = (float*)d_ws;

    k_zero<<<(WS_ZERON + 255) / 256, 256, 0, stream>>>(ws);
    k_class_sums<<<512, 256, 0, stream>>>(xT, lab, ws);
    k_stats<<<1, 256, 0, stream>>>(ws);
    k_intra<<<256, 256, 0, stream>>>(xT, lab, ws);
    k_topk_gather<<<NN, 256, 0, stream>>>(x, out, ws);
}